// DecoderTreeNN_21844203668324
// MI455X (gfx1250) — compile-verified
//
#include <hip/hip_runtime.h>
#include <hip/hip_bf16.h>

// ---- problem constants ----
#define EE   256
#define VV   32000
#define BB   32
#define MM   2048
#define NTT  64
#define NNN  64
#define ECH  64
#define NNODE 65

typedef __attribute__((ext_vector_type(2))) float v2f;
typedef __attribute__((ext_vector_type(8))) float v8f;

// ---------------------------------------------------------------------------
// Kernel 1: fused KB gather + type multiply + 3-step tree scatter-add in LDS.
// One block per (b,t) pair and E-chunk of 64. Outputs roots (B,NT,E).
// ---------------------------------------------------------------------------
__global__ __launch_bounds__(256) void k_tree(
    const int* __restrict__ kb_values,   // B*NT*NN*4
    const int* __restrict__ kb_types,    // B*NT*NN
    const int* __restrict__ kb_fathers,  // B*NT*NN
    const int* __restrict__ kb_nl,       // B*NT*NN
    const float* __restrict__ C0,        // V*E  (C_emb[0])
    const float* __restrict__ T_emb,     // 8*E
    float* __restrict__ roots)           // B*NT*E
{
    __shared__ float ne_s[NNODE][ECH];
    __shared__ float nxt_s[NNODE][ECH];
    __shared__ int   fa_s[NNN];
    __shared__ int   nl_s[NNN];

    const int bt  = blockIdx.x;          // b*NT + t
    const int e0  = blockIdx.y * ECH;
    const int tid = threadIdx.x;
    const int e   = tid & 63;
    const int ng  = tid >> 6;            // 0..3

    if (tid < NNN) {
        fa_s[tid] = kb_fathers[(size_t)bt * NNN + tid];
        nl_s[tid] = 2 - kb_nl[(size_t)bt * NNN + tid]; // COMP_STEP-1 - n_layers
    }

    for (int n = ng; n < NNN; n += 4) {
        const int* vi = kb_values + ((size_t)bt * NNN + n) * 4;
        float s = 0.f;
#pragma unroll
        for (int l = 0; l < 4; ++l)
            s += C0[(size_t)vi[l] * EE + e0 + e];
        int ty = kb_types[(size_t)bt * NNN + n];
        ne_s[n][e] = s * T_emb[ty * EE + e0 + e];
    }
    if (ng == 0) ne_s[NNN][e] = 0.f;     // appended zero node
    __syncthreads();

    for (int step = 0; step < 3; ++step) {
        for (int n = ng; n < NNODE; n += 4) nxt_s[n][e] = 0.f;
        __syncthreads();
        for (int n = ng; n < NNN; n += 4)
            atomicAdd(&nxt_s[fa_s[n]][e], ne_s[n][e]);   // ds_add_f32
        __syncthreads();
        for (int n = ng; n < NNN; n += 4)
            if (nl_s[n] == step) ne_s[n][e] += nxt_s[n][e];
        __syncthreads();
    }
    if (ng == 0) roots[(size_t)bt * EE + e0 + e] = ne_s[0][e];
}

// ---------------------------------------------------------------------------
// Kernel 2: attention (via algebraic collapse of Wk/Wv GEMMs) + GRU cell.
// One block of 256 threads per batch element. Outputs cur_state and u0.
// ---------------------------------------------------------------------------
__global__ __launch_bounds__(256) void k_attn_gru(
    const int* __restrict__ dec_in,
    const float* __restrict__ hidden,    // B*E
    const float* __restrict__ C0,
    const float* __restrict__ roots,     // B*NT*E
    const float* __restrict__ Wq, const float* __restrict__ Wk,
    const float* __restrict__ Wv,
    const float* __restrict__ W_ih, const float* __restrict__ W_hh,
    const float* __restrict__ b_ih, const float* __restrict__ b_hh,
    float* __restrict__ cur_out,         // B*E
    float* __restrict__ u0)              // B*E
{
    const int b = blockIdx.x;
    const int e = threadIdx.x;
    __shared__ float h_s[EE], x_s[EE], q_s[EE], kq_s[EE], rs_s[EE];
    __shared__ float w_s[NTT];

    h_s[e] = hidden[b * EE + e];
    x_s[e] = C0[(size_t)dec_in[b] * EE + e];
    __syncthreads();

    float q = 0.f;
    for (int j = 0; j < EE; ++j) q += Wq[e * EE + j] * h_s[j];   // query = h @ Wq.T
    q_s[e] = q;
    __syncthreads();

    float kq = 0.f;
    for (int i = 0; i < EE; ++i) kq += Wk[i * EE + e] * q_s[i];  // kq = Wk^T q
    kq_s[e] = kq;
    __syncthreads();

    if (e < NTT) {
        const float* rr = roots + ((size_t)b * NTT + e) * EE;
        float lg = 0.f, sr = 0.f;
        for (int j = 0; j < EE; ++j) { float r = rr[j]; lg += r * kq_s[j]; sr += r; }
        if (sr == 0.f) lg += -1e9f;
        w_s[e] = lg;
    }
    __syncthreads();
    if (e == 0) {
        float mx = w_s[0];
        for (int n = 1; n < NTT; ++n) mx = fmaxf(mx, w_s[n]);
        float sum = 0.f;
        for (int n = 0; n < NTT; ++n) { float v = expf(w_s[n] - mx); w_s[n] = v; sum += v; }
        float inv = 1.f / sum;
        for (int n = 0; n < NTT; ++n) w_s[n] *= inv;
    }
    __syncthreads();

    float rs = 0.f;
    for (int n = 0; n < NTT; ++n) rs += w_s[n] * roots[((size_t)b * NTT + n) * EE + e];
    rs_s[e] = rs;
    __syncthreads();

    float af = 0.f;
    for (int j = 0; j < EE; ++j) af += Wv[e * EE + j] * rs_s[j]; // attn_feat = Wv rsum

    float gi0 = b_ih[e], gi1 = b_ih[EE + e], gi2 = b_ih[2 * EE + e];
    float gh0 = b_hh[e], gh1 = b_hh[EE + e], gh2 = b_hh[2 * EE + e];
    for (int j = 0; j < EE; ++j) {
        float x = x_s[j], h = h_s[j];
        gi0 += W_ih[e * EE + j] * x;
        gi1 += W_ih[(EE + e) * EE + j] * x;
        gi2 += W_ih[(2 * EE + e) * EE + j] * x;
        gh0 += W_hh[e * EE + j] * h;
        gh1 += W_hh[(EE + e) * EE + j] * h;
        gh2 += W_hh[(2 * EE + e) * EE + j] * h;
    }
    float r  = 1.f / (1.f + expf(-(gi0 + gh0)));
    float z  = 1.f / (1.f + expf(-(gi1 + gh1)));
    float nn = tanhf(gi2 + r * gh2);
    float hnew = (1.f - z) * nn + z * h_s[e];
    float cur = hnew + af;
    cur_out[b * EE + e] = cur;
    u0[b * EE + e] = cur;
}

// ---------------------------------------------------------------------------
// Kernel 3: prob_lg[b,m] = (sum_l C[h][story[b,m,l]]) . u[b]
// One wave per (b,m); gathers are 128B coalesced and hit L2 (C table 33MB).
// ---------------------------------------------------------------------------
__global__ __launch_bounds__(256) void k_hoplg(
    const int* __restrict__ story,       // B*M*4
    const float* __restrict__ Ch,        // C_emb[h] base
    const float* __restrict__ u,         // B*E
    float* __restrict__ lg)              // B*M
{
    const int b    = blockIdx.y;
    const int m    = blockIdx.x * 8 + (threadIdx.x >> 5);
    const int lane = threadIdx.x & 31;
    __shared__ float u_s[EE];
    if (threadIdx.x < EE) u_s[threadIdx.x] = u[b * EE + threadIdx.x];
    __syncthreads();

    const int* st = story + ((size_t)b * MM + m) * 4;
    float acc = 0.f;
#pragma unroll
    for (int l = 0; l < 4; ++l) {
        size_t base = (size_t)st[l] * EE;
#pragma unroll
        for (int d = 0; d < 8; ++d) {
            int j = lane + d * 32;
            acc += Ch[base + j] * u_s[j];
        }
    }
    for (int off = 16; off; off >>= 1) acc += __shfl_xor(acc, off, 32);
    if (lane == 0) lg[(size_t)b * MM + m] = acc;
}

// ---------------------------------------------------------------------------
// Kernel 4: softmax stats (max, sumexp) over M per batch element.
// ---------------------------------------------------------------------------
__global__ __launch_bounds__(256) void k_stats(
    const float* __restrict__ lg, float* __restrict__ stats)
{
    const int b = blockIdx.x, tid = threadIdx.x;
    __shared__ float red[256];
    float mx = -1e30f;
    for (int m = tid; m < MM; m += 256) mx = fmaxf(mx, lg[(size_t)b * MM + m]);
    red[tid] = mx; __syncthreads();
    for (int s = 128; s; s >>= 1) { if (tid < s) red[tid] = fmaxf(red[tid], red[tid + s]); __syncthreads(); }
    mx = red[0]; __syncthreads();
    float sum = 0.f;
    for (int m = tid; m < MM; m += 256) sum += expf(lg[(size_t)b * MM + m] - mx);
    red[tid] = sum; __syncthreads();
    for (int s = 128; s; s >>= 1) { if (tid < s) red[tid] += red[tid + s]; __syncthreads(); }
    if (tid == 0) { stats[b * 2] = mx; stats[b * 2 + 1] = red[0]; }
}

// ---------------------------------------------------------------------------
// Kernel 5: o_k[b,:] += sum_m softmax(lg)[b,m] * (sum_l C[h+1][story[b,m,l]])
// Block = 256 threads (one per e-channel), 128 m's per block, atomic finish.
// ---------------------------------------------------------------------------
__global__ __launch_bounds__(256) void k_ok(
    const int* __restrict__ story,
    const float* __restrict__ Chn,       // C_emb[h+1]
    const float* __restrict__ lg,
    const float* __restrict__ stats,
    float* __restrict__ ok)              // B*E (pre-zeroed)
{
    const int b  = blockIdx.y;
    const int m0 = blockIdx.x * 128;
    const int e  = threadIdx.x;
    const float mx  = stats[b * 2];
    const float inv = 1.f / stats[b * 2 + 1];
    float acc = 0.f;
    for (int mi = 0; mi < 128; ++mi) {
        int m = m0 + mi;
        float w = expf(lg[(size_t)b * MM + m] - mx) * inv;
        const int* st = story + ((size_t)b * MM + m) * 4;
        float s = Chn[(size_t)st[0] * EE + e] + Chn[(size_t)st[1] * EE + e]
                + Chn[(size_t)st[2] * EE + e] + Chn[(size_t)st[3] * EE + e];
        acc += w * s;
    }
    atomicAdd(&ok[b * EE + e], acc);
}

// ---------------------------------------------------------------------------
// Kernel 6: u_next = u_prev + ok ; optionally pack ucat = [u_prev | ok]
// ---------------------------------------------------------------------------
__global__ __launch_bounds__(256) void k_uprep(
    const float* __restrict__ u_prev, const float* __restrict__ okv,
    float* __restrict__ u_next, float* __restrict__ ucat)
{
    int i = blockIdx.x * 256 + threadIdx.x;          // over B*E
    int b = i >> 8, e = i & 255;
    float a = u_prev[i], o = okv[i];
    if (u_next) u_next[i] = a + o;
    if (ucat) { ucat[b * 512 + e] = a; ucat[b * 512 + 256 + e] = o; }
}

// ---------------------------------------------------------------------------
// Kernel 7: p_vocab = [u0|ok0](32x512) @ W1_w(32000x512)^T + b1, via
// V_WMMA_F32_16X16X4_F32.  One wave per 16-wide vocab tile, both 16-row
// b-tiles share the B fragment so W1_w streams exactly once (65MB, HBM-bound).
// A/B fragment layout per ISA 7.12.2: VGPR0/1 = K{0,1} lanes 0-15, K{2,3}
// lanes 16-31; M (or N) = lane%16.
// ---------------------------------------------------------------------------
__global__ __launch_bounds__(256) void k_pv(
    const float* __restrict__ ucat,      // 32 x 512
    const float* __restrict__ W1,        // 32000 x 512
    const float* __restrict__ b1,        // 32000
    float* __restrict__ out)             // 32 x 32000
{
    const int wave = threadIdx.x >> 5;
    const int lane = threadIdx.x & 31;
    const int vt   = blockIdx.x * 8 + wave;   // vocab tile 0..1999
    const int vbase = vt * 16;
    const int half = lane >> 4;
    const int lr   = lane & 15;

    v8f c0 = {}; v8f c1 = {};
    const float* Arow0 = ucat + (size_t)lr * 512 + half * 2;
    const float* Arow1 = ucat + (size_t)(16 + lr) * 512 + half * 2;
    const float* Brow  = W1 + (size_t)(vbase + lr) * 512 + half * 2;

    for (int k = 0; k < 512; k += 4) {
        v2f a0 = *(const v2f*)(Arow0 + k);
        v2f a1 = *(const v2f*)(Arow1 + k);
        v2f bb = *(const v2f*)(Brow + k);
        c0 = __builtin_amdgcn_wmma_f32_16x16x4_f32(false, a0, false, bb,
                                                   (short)0, c0, false, false);
        c1 = __builtin_amdgcn_wmma_f32_16x16x4_f32(false, a1, false, bb,
                                                   (short)0, c1, false, false);
    }

    const float bias = b1[vbase + lr];
    const int n = vbase + lr;
#pragma unroll
    for (int r = 0; r < 8; ++r) {
        int m0 = r + half * 8;                       // D layout: M=r (+8 hi half)
        out[(size_t)m0 * VV + n]        = c0[r] + bias;
        out[(size_t)(m0 + 16) * VV + n] = c1[r] + bias;
    }
}

// ---------------------------------------------------------------------------
extern "C" void kernel_launch(void* const* d_in, const int* in_sizes, int n_in,
                              void* d_out, int out_size, void* d_ws, size_t ws_size,
                              hipStream_t stream) {
    const int*   dec       = (const int*)d_in[0];
    const int*   story     = (const int*)d_in[1];
    const int*   kb_values = (const int*)d_in[2];
    const int*   kb_types  = (const int*)d_in[3];
    const int*   kb_fath   = (const int*)d_in[4];
    const int*   kb_nl     = (const int*)d_in[5];
    const float* hidden    = (const float*)d_in[6];
    const float* C_emb     = (const float*)d_in[7];   // (4, V, E)
    const float* T_emb     = (const float*)d_in[8];
    const float* Wq        = (const float*)d_in[9];
    const float* Wk        = (const float*)d_in[10];
    const float* Wv        = (const float*)d_in[11];
    const float* W1_w      = (const float*)d_in[12];
    const float* W1_b      = (const float*)d_in[13];
    const float* W_ih      = (const float*)d_in[14];
    const float* W_hh      = (const float*)d_in[15];
    const float* b_ih      = (const float*)d_in[16];
    const float* b_hh      = (const float*)d_in[17];

    float* out       = (float*)d_out;
    float* p_ptr     = out;                           // B*M
    float* p_vocab   = out + (size_t)BB * MM;         // B*V
    float* cur_state = out + (size_t)BB * MM + (size_t)BB * VV; // B*E

    float* ws    = (float*)d_ws;
    float* roots = ws;                                // B*NT*E
    float* u0    = roots + (size_t)BB * NTT * EE;
    float* u1    = u0 + BB * EE;
    float* u2    = u1 + BB * EE;
    float* ucat  = u2 + BB * EE;                      // B*512
    float* lg0   = ucat + BB * 512;                   // B*M
    float* lg1   = lg0 + (size_t)BB * MM;
    float* ok0   = lg1 + (size_t)BB * MM;
    float* ok1   = ok0 + BB * EE;
    float* st0   = ok1 + BB * EE;                     // B*2
    float* st1   = st0 + BB * 2;

    const size_t CHOP = (size_t)VV * EE;              // C_emb hop stride

    // 1) tree -> roots
    k_tree<<<dim3(BB * NTT, EE / ECH), 256, 0, stream>>>(
        kb_values, kb_types, kb_fath, kb_nl, C_emb, T_emb, roots);

    // 2) attention + GRU -> cur_state, u0
    k_attn_gru<<<BB, 256, 0, stream>>>(dec, hidden, C_emb, roots,
        Wq, Wk, Wv, W_ih, W_hh, b_ih, b_hh, cur_state, u0);

    // hop 0
    k_hoplg<<<dim3(MM / 8, BB), 256, 0, stream>>>(story, C_emb, u0, lg0);
    k_stats<<<BB, 256, 0, stream>>>(lg0, st0);
    hipMemsetAsync(ok0, 0, (size_t)BB * EE * sizeof(float), stream);
    k_ok<<<dim3(MM / 128, BB), 256, 0, stream>>>(story, C_emb + CHOP, lg0, st0, ok0);
    k_uprep<<<(BB * EE) / 256, 256, 0, stream>>>(u0, ok0, u1, ucat);

    // p_vocab GEMM (WMMA f32)
    k_pv<<<(VV / 16) / 8, 256, 0, stream>>>(ucat, W1_w, W1_b, p_vocab);

    // hop 1
    k_hoplg<<<dim3(MM / 8, BB), 256, 0, stream>>>(story, C_emb + CHOP, u1, lg1);
    k_stats<<<BB, 256, 0, stream>>>(lg1, st1);
    hipMemsetAsync(ok1, 0, (size_t)BB * EE * sizeof(float), stream);
    k_ok<<<dim3(MM / 128, BB), 256, 0, stream>>>(story, C_emb + 2 * CHOP, lg1, st1, ok1);
    k_uprep<<<(BB * EE) / 256, 256, 0, stream>>>(u1, ok1, u2, nullptr);

    // hop 2: p_ptr = prob_lg (logits of final hop); o_k/u of hop 2 are dead.
    k_hoplg<<<dim3(MM / 8, BB), 256, 0, stream>>>(story, C_emb + 2 * CHOP, u2, p_ptr);
}